// EdgeConv_52441550684778
// MI455X (gfx1250) — compile-verified
//
#include <hip/hip_runtime.h>

#define N_PTS 20000
#define KNBR  16
#define DX    64
#define E_TOT (N_PTS * KNBR)
#define RBLK  (N_PTS / 16)   // 1250 row blocks of 16
#define KNN_WPB 5            // waves per block in knn kernel (250 blocks * 5 = 1250)
#define TROW 65              // padded LDS row stride (64 + 1 pad dword from TDM)

typedef __attribute__((ext_vector_type(2))) float v2f;
typedef __attribute__((ext_vector_type(8))) float v8f;
typedef __attribute__((ext_vector_type(4))) unsigned int v4u;
typedef __attribute__((ext_vector_type(8))) int v8i;
typedef __attribute__((ext_vector_type(4))) int v4i;

static __device__ __forceinline__ v8f wmma_f32x4(v2f a, v2f b, v8f c) {
  // D = A(16x4,f32) * B(4x16,f32) + C(16x16,f32)
  return __builtin_amdgcn_wmma_f32_16x16x4_f32(
      /*neg_a=*/false, a, /*neg_b=*/false, b,
      /*c_mod=*/(short)0, c, /*reuse_a=*/false, /*reuse_b=*/false);
}

static __device__ __forceinline__ void lds_wait() {
  asm volatile("s_wait_dscnt 0" ::: "memory");
}

// TDM: DMA one 16x64 f32 tile (row-major, row stride 64) from global into LDS,
// padding +1 dword after every 64 dwords (-> 65-dword LDS row stride, bank-safe).
// D# layout per CDNA5 ISA ch.8 (group0/1; groups 2-4 unused for 2D -> zeros).
static __device__ __forceinline__ void tdm_load_tile(const float* gsrc, unsigned lds_off) {
  unsigned long long ga = (unsigned long long)(uintptr_t)gsrc;
  v4u g0;
  g0[0] = 1u;                                    // count=1, user descriptor
  g0[1] = lds_off;                               // lds_addr (bytes)
  g0[2] = (unsigned)ga;                          // global_addr[31:0]
  g0[3] = (unsigned)((ga >> 32) & 0x01FFFFFFu)   // global_addr[56:32]
        | (2u << 30);                            // type = 2 ("image")
  v8i g1;
  g1[0] = (int)((2u << 16)      // data_size = 4 bytes
              | (1u << 20)      // pad_enable
              | (5u << 22)      // pad_interval: 2^(5+1)=64 dwords
              | (0u << 25));    // pad_amount: 0 -> 1 dword
  g1[1] = (int)(64u << 16);     // tensor_dim0 = 64 (low16 at [31:16])
  g1[2] = (int)(16u << 16);     // dim0 hi16=0 | tensor_dim1 = 16 (low16)
  g1[3] = (int)(64u << 16);     // dim1 hi16=0 | tile_dim0 = 64
  g1[4] = (int)16;              // tile_dim1 = 16, tile_dim2 = 0
  g1[5] = (int)64;              // tensor_dim0_stride = 64 (low32)
  g1[6] = 0;                    // stride0 hi16 = 0, stride1 lo16 = 0
  g1[7] = 0;
  v4i gz4 = (v4i){0, 0, 0, 0};
  v8i gz8 = (v8i){0, 0, 0, 0, 0, 0, 0, 0};
  // 6-arg form on this toolchain: (g0, g1, g2, g3, g4, cpol)
  __builtin_amdgcn_tensor_load_to_lds(g0, g1, gz4, gz4, gz8, 0);
}

// ---------------------------------------------------------------------------
// Kernel 0: W2 = phi_W - theta_W ; biasSum = theta_b + phi_b
// ---------------------------------------------------------------------------
__global__ void prep_kernel(const float* __restrict__ thW, const float* __restrict__ phW,
                            const float* __restrict__ thb, const float* __restrict__ phb,
                            float* __restrict__ W2, float* __restrict__ biasSum) {
  int i = blockIdx.x * blockDim.x + threadIdx.x;
  if (i < DX * DX) W2[i] = phW[i] - thW[i];
  if (i < DX)      biasSum[i] = thb[i] + phb[i];
}

// ---------------------------------------------------------------------------
// Kernel 1: A = x @ theta_W ; B = x @ (phi_W - theta_W)    (WMMA f32 GEMM)
// One wave per 16-row block.
// ---------------------------------------------------------------------------
__global__ void __launch_bounds__(256)
transform_kernel(const float* __restrict__ x, const float* __restrict__ thW,
                 const float* __restrict__ W2,
                 float* __restrict__ Aout, float* __restrict__ Bout) {
  int wave = (int)(blockIdx.x * (blockDim.x >> 5)) + (int)(threadIdx.x >> 5);
  int lane = (int)(threadIdx.x & 31);
  if (wave >= RBLK) return;
  int r0   = wave * 16;
  int m    = lane & 15;
  int half = lane >> 4;
  int k0   = half * 2;
  int n    = m;

  v8f accA[4], accB[4];
#pragma unroll
  for (int t = 0; t < 4; ++t) {
    accA[t] = (v8f){0.f,0.f,0.f,0.f,0.f,0.f,0.f,0.f};
    accB[t] = (v8f){0.f,0.f,0.f,0.f,0.f,0.f,0.f,0.f};
  }

#pragma unroll 4
  for (int kk = 0; kk < 16; ++kk) {
    int kb = kk * 4 + k0;
    v2f a;
    a.x = x[(r0 + m) * DX + kb];
    a.y = x[(r0 + m) * DX + kb + 1];
#pragma unroll
    for (int t = 0; t < 4; ++t) {
      int col = t * 16 + n;
      v2f bt, bw;
      bt.x = thW[kb * DX + col];
      bt.y = thW[(kb + 1) * DX + col];
      bw.x = W2[kb * DX + col];
      bw.y = W2[(kb + 1) * DX + col];
      accA[t] = wmma_f32x4(a, bt, accA[t]);
      accB[t] = wmma_f32x4(a, bw, accB[t]);
    }
  }

#pragma unroll
  for (int t = 0; t < 4; ++t)
#pragma unroll
    for (int c = 0; c < 8; ++c) {
      int row = r0 + c + 8 * half;
      Aout[row * DX + t * 16 + n] = accA[t][c];
      Bout[row * DX + t * 16 + n] = accB[t][c];
    }
}

// ---------------------------------------------------------------------------
// Kernel 2: x_new[i] = A[i] + biasSum + max_k B[src[i*16+k]] ; x2[i] = |x_new|^2
// One wave per node (64 dims -> 2 per lane).
// ---------------------------------------------------------------------------
__global__ void __launch_bounds__(256)
aggregate_kernel(const float* __restrict__ A, const float* __restrict__ B,
                 const float* __restrict__ biasSum, const int* __restrict__ src,
                 float* __restrict__ xnew, float* __restrict__ x2) {
  int wave = (int)(blockIdx.x * (blockDim.x >> 5)) + (int)(threadIdx.x >> 5);
  int lane = (int)(threadIdx.x & 31);
  if (wave >= N_PTS) return;

  float m0 = -3.4e38f, m1 = -3.4e38f;
  const int* sp = src + wave * KNBR;
#pragma unroll 4
  for (int k = 0; k < KNBR; ++k) {
    int s = sp[k];
    m0 = fmaxf(m0, B[s * DX + lane]);
    m1 = fmaxf(m1, B[s * DX + 32 + lane]);
  }
  float v0 = A[wave * DX + lane]      + biasSum[lane]      + m0;
  float v1 = A[wave * DX + 32 + lane] + biasSum[32 + lane] + m1;
  xnew[wave * DX + lane]      = v0;
  xnew[wave * DX + 32 + lane] = v1;

  float ss = v0 * v0 + v1 * v1;
#pragma unroll
  for (int off = 16; off > 0; off >>= 1) ss += __shfl_xor(ss, off, 32);
  if (lane == 0) x2[wave] = ss;
}

// ---------------------------------------------------------------------------
// Kernel 3: en path. Tiny MLPs 4->4->5->4->4, mean over 16 in-edges.
// ---------------------------------------------------------------------------
static __device__ __forceinline__ void mlp4(const float in[4], float out[4],
    const float* __restrict__ W0, const float* __restrict__ b0,
    const float* __restrict__ W1, const float* __restrict__ b1,
    const float* __restrict__ W2, const float* __restrict__ b2,
    const float* __restrict__ W3, const float* __restrict__ b3) {
  float h0[4], h1[5], h2[4];
#pragma unroll
  for (int j = 0; j < 4; ++j) {
    float s = b0[j];
#pragma unroll
    for (int i = 0; i < 4; ++i) s += in[i] * W0[i * 4 + j];
    h0[j] = fmaxf(s, 0.f);
  }
#pragma unroll
  for (int j = 0; j < 5; ++j) {
    float s = b1[j];
#pragma unroll
    for (int i = 0; i < 4; ++i) s += h0[i] * W1[i * 5 + j];
    h1[j] = fmaxf(s, 0.f);
  }
#pragma unroll
  for (int j = 0; j < 4; ++j) {
    float s = b2[j];
#pragma unroll
    for (int i = 0; i < 5; ++i) s += h1[i] * W2[i * 4 + j];
    h2[j] = fmaxf(s, 0.f);
  }
#pragma unroll
  for (int j = 0; j < 4; ++j) {
    float s = b3[j];
#pragma unroll
    for (int i = 0; i < 4; ++i) s += h2[i] * W3[i * 4 + j];
    out[j] = s;
  }
}

__global__ void __launch_bounds__(256)
en_kernel(const float* __restrict__ en, const int* __restrict__ src,
          const float* tW0, const float* tb0, const float* tW1, const float* tb1,
          const float* tW2, const float* tb2, const float* tW3, const float* tb3,
          const float* pW0, const float* pb0, const float* pW1, const float* pb1,
          const float* pW2, const float* pb2, const float* pW3, const float* pb3,
          float* __restrict__ enout) {
  int i = blockIdx.x * blockDim.x + threadIdx.x;
  if (i >= N_PTS) return;
  float ei[4];
#pragma unroll
  for (int d = 0; d < 4; ++d) ei[d] = en[i * 4 + d];
  float acc[4] = {0.f, 0.f, 0.f, 0.f};
  for (int k = 0; k < KNBR; ++k) {
    int s = src[i * KNBR + k];
    float es[4], df[4];
#pragma unroll
    for (int d = 0; d < 4; ++d) { es[d] = en[s * 4 + d]; df[d] = ei[d] - es[d]; }
    float o1[4], o2[4];
    mlp4(df, o1, tW0, tb0, tW1, tb1, tW2, tb2, tW3, tb3);
    mlp4(es, o2, pW0, pb0, pW1, pb1, pW2, pb2, pW3, pb3);
#pragma unroll
    for (int d = 0; d < 4; ++d) acc[d] += o1[d] + o2[d];
  }
#pragma unroll
  for (int d = 0; d < 4; ++d) enout[i * 4 + d] = acc[d] * (1.0f / 16.0f);
}

// ---------------------------------------------------------------------------
// Kernel 4: fused KNN. 5 waves/block, each owns a 16-row block; the block
// sweeps the 1250 column tiles together. Wave 0 DMAs each 16x64 tile into a
// double-buffered LDS slot via the Tensor Data Mover (padded rows -> stride 65
// dwords, bank-conflict-free ds_load_b64 fragment reads); all waves consume it
// with f32 WMMA, rank by key = x2[col] - 2*dot, keep per-lane sorted top-16,
// and merge half-lists through LDS at the end.
// ---------------------------------------------------------------------------
__global__ void __launch_bounds__(32 * KNN_WPB)
knn_kernel(const float* __restrict__ xn, const float* __restrict__ x2,
           float* __restrict__ out_src, float* __restrict__ out_dst) {
  __shared__ float s_btile[2][1056];            // TDM dest: 16 rows * 65 dw (padded)
  __shared__ float s_tile[KNN_WPB][16][17];     // per-wave key-tile transpose
  __shared__ float s_md[KNN_WPB][32][16];       // merge: distances
  __shared__ int   s_mi[KNN_WPB][32][16];       // merge: indices

  int wv   = (int)(threadIdx.x >> 5);
  int wave = (int)blockIdx.x * KNN_WPB + wv;    // exact: 250 * 5 = 1250
  int lane = (int)(threadIdx.x & 31);

  int r0   = wave * 16;
  int m    = lane & 15;
  int half = lane >> 4;
  int k0   = half * 2;
  int n    = m;

  // Preload A-fragments for this wave's 16 rows (16x64 tile in registers).
  v2f af[16];
#pragma unroll
  for (int kk = 0; kk < 16; ++kk) {
    int kb = kk * 4 + k0;
    af[kk].x = xn[(r0 + m) * DX + kb];
    af[kk].y = xn[(r0 + m) * DX + kb + 1];
  }

  float bd[16];
  int   bi[16];
#pragma unroll
  for (int t = 0; t < 16; ++t) { bd[t] = 3.4e38f; bi[t] = 0; }

  int row_local = m;     // row this lane selects for
  int cb = half * 8;     // column-half of the tile this lane scans

  // Prologue: DMA tile 0 into slot 0.
  if (wv == 0)
    tdm_load_tile(xn, (unsigned)(uintptr_t)(void*)&s_btile[0][0]);

  for (int tj = 0; tj < RBLK; ++tj) {
    int slot = tj & 1;
    if (wv == 0) __builtin_amdgcn_s_wait_tensorcnt(0);   // tile tj landed
    __syncthreads();   // publish tile tj; also proves slot (tj+1)&1 fully read
    if (wv == 0 && tj + 1 < RBLK)
      tdm_load_tile(xn + (size_t)(tj + 1) * 16 * DX,
                    (unsigned)(uintptr_t)(void*)&s_btile[(tj + 1) & 1][0]);

    int c0 = tj * 16;
    v8f acc = (v8f){0.f,0.f,0.f,0.f,0.f,0.f,0.f,0.f};
    const float* brow = &s_btile[slot][(unsigned)n * TROW + k0];
#pragma unroll
    for (int kk = 0; kk < 16; ++kk) {
      v2f b;
      b.x = brow[kk * 4];
      b.y = brow[kk * 4 + 1];
      acc = wmma_f32x4(af[kk], b, acc);
    }
    // key[M][N] = x2[col] - 2*dot (x2[row] omitted: constant per row)
    float cx2 = x2[c0 + n];
#pragma unroll
    for (int c = 0; c < 8; ++c)
      s_tile[wv][c + 8 * half][n] = cx2 - 2.0f * acc[c];
    lds_wait();  // intra-wave LDS write->read ordering

#pragma unroll
    for (int j = 0; j < 8; ++j) {
      float d = s_tile[wv][row_local][cb + j];
      int  jg = c0 + cb + j;
      if (d < bd[15]) {
        bd[15] = d; bi[15] = jg;
#pragma unroll
        for (int t = 15; t > 0; --t) {
          if (bd[t] < bd[t - 1]) {
            float td = bd[t]; bd[t] = bd[t - 1]; bd[t - 1] = td;
            int   ti = bi[t]; bi[t] = bi[t - 1]; bi[t - 1] = ti;
          }
        }
      }
    }
    lds_wait();  // s_tile reads done before next tile's writes
  }

  // Merge the two per-row half-lists (lane L with lane L+16) via LDS.
#pragma unroll
  for (int t = 0; t < 16; ++t) { s_md[wv][lane][t] = bd[t]; s_mi[wv][lane][t] = bi[t]; }
  lds_wait();

  if (half == 0) {
    int ia = 0, ib = 0;
    int row = r0 + lane;
    float frow = (float)row;
    for (int t = 0; t < 16; ++t) {
      float da = s_md[wv][lane][ia];
      float db = s_md[wv][lane + 16][ib];
      int id;
      if (da <= db) { id = s_mi[wv][lane][ia]; ++ia; }
      else          { id = s_mi[wv][lane + 16][ib]; ++ib; }
      out_src[row * KNBR + t] = (float)id;  // N < 2^24: exact in f32
      out_dst[row * KNBR + t] = frow;
    }
  }
}

// ---------------------------------------------------------------------------
extern "C" void kernel_launch(void* const* d_in, const int* in_sizes, int n_in,
                              void* d_out, int out_size, void* d_ws, size_t ws_size,
                              hipStream_t stream) {
  (void)in_sizes; (void)n_in; (void)out_size; (void)ws_size;

  const float* x    = (const float*)d_in[0];
  const float* en   = (const float*)d_in[1];
  const int*   src  = (const int*)d_in[2];
  /* d_in[3] = dst: structurally repeat(arange(N),16); not needed */
  const float* thW  = (const float*)d_in[4];
  const float* thb  = (const float*)d_in[5];
  const float* phW  = (const float*)d_in[6];
  const float* phb  = (const float*)d_in[7];
  const float* tW0 = (const float*)d_in[8],  *tb0 = (const float*)d_in[9];
  const float* tW1 = (const float*)d_in[10], *tb1 = (const float*)d_in[11];
  const float* tW2 = (const float*)d_in[12], *tb2 = (const float*)d_in[13];
  const float* tW3 = (const float*)d_in[14], *tb3 = (const float*)d_in[15];
  const float* pW0 = (const float*)d_in[16], *pb0 = (const float*)d_in[17];
  const float* pW1 = (const float*)d_in[18], *pb1 = (const float*)d_in[19];
  const float* pW2 = (const float*)d_in[20], *pb2 = (const float*)d_in[21];
  const float* pW3 = (const float*)d_in[22], *pb3 = (const float*)d_in[23];

  float* ws      = (float*)d_ws;
  float* W2      = ws;                       // 4096
  float* biasSum = ws + DX * DX;             // 64
  float* A_ws    = biasSum + DX;             // N*64
  float* B_ws    = A_ws + (size_t)N_PTS * DX;// N*64
  float* x2      = B_ws + (size_t)N_PTS * DX;// N

  float* xnew  = (float*)d_out;              // N*64
  float* enout = xnew + (size_t)N_PTS * DX;  // N*4
  float* osrc  = enout + (size_t)N_PTS * 4;  // N*16
  float* odst  = osrc + (size_t)N_PTS * KNBR;// N*16

  prep_kernel<<<16, 256, 0, stream>>>(thW, phW, thb, phb, W2, biasSum);

  transform_kernel<<<(RBLK + 7) / 8, 256, 0, stream>>>(x, thW, W2, A_ws, B_ws);

  aggregate_kernel<<<(N_PTS + 7) / 8, 256, 0, stream>>>(A_ws, B_ws, biasSum, src, xnew, x2);

  en_kernel<<<(N_PTS + 255) / 256, 256, 0, stream>>>(
      en, src,
      tW0, tb0, tW1, tb1, tW2, tb2, tW3, tb3,
      pW0, pb0, pW1, pb1, pW2, pb2, pW3, pb3,
      enout);

  knn_kernel<<<RBLK / KNN_WPB, 32 * KNN_WPB, 0, stream>>>(xnew, x2, osrc, odst);
}